// test_gru_936302870994
// MI455X (gfx1250) — compile-verified
//
#include <hip/hip_runtime.h>

// GRU, H=3, T=2048, B=4096, recurrent chain of T steps.
// One wave32 owns 16 batch columns. h-side projection = one
// V_WMMA_F32_16X16X4_F32 per step. Hidden-state ownership is split across
// lane halves to match the WMMA B-operand layout exactly:
//   lanes 0-15  : h0,h1 of batch b=lane      (B operand K0/K1, D rows 0-7)
//   lanes 16-31 : h2    of batch b=lane-16   (B operand K2,    D row 8)
// -> zero shuffles to rebuild the B operand, gate-8 consumed in-lane; the
// only cross-half hop per step is {r2,z2}, done with v_permlane16_swap_b32
// (pure VALU, no LDS roundtrip / dscnt wait) when available.

typedef float v2f __attribute__((ext_vector_type(2)));
typedef float v8f __attribute__((ext_vector_type(8)));
typedef unsigned int v2u __attribute__((ext_vector_type(2)));

#define HH 3
#define GG 9   // 3*H gates

__device__ __forceinline__ float fast_sigmoid(float x) {
    float e = __builtin_amdgcn_exp2f(-x * 1.4426950408889634f);
    return __builtin_amdgcn_rcpf(1.0f + e);
}

__device__ __forceinline__ float fast_tanh(float x) {
    // 1 - 2/(exp(2x)+1): saturates cleanly to +/-1 at extremes
    float e = __builtin_amdgcn_exp2f(x * 2.8853900817779268f);
    return 1.0f - 2.0f * __builtin_amdgcn_rcpf(e + 1.0f);
}

// Give lanes 16-31 the value held by lanes 0-15 (lane ^ 16 source).
// v_permlane16_swap_b32 with vdst_old == src0 == v returns, in result[0]:
//   row0 (lanes 0-15)  = vdst_old.row0 = v (unchanged)
//   row1 (lanes 16-31) = src0.row0    = v from lanes 0-15   <- what we need
__device__ __forceinline__ float bcast_lo_to_hi(float v) {
#if defined(__has_builtin) && __has_builtin(__builtin_amdgcn_permlane16_swap)
    unsigned u = __builtin_bit_cast(unsigned, v);
    v2u r = __builtin_amdgcn_permlane16_swap(u, u, false, false);
    return __builtin_bit_cast(float, r.x);
#else
    return __shfl_xor(v, 16, 32);   // ds_bpermute fallback
#endif
}

__global__ __launch_bounds__(128) void gru_wmma_kernel(
    const float* __restrict__ input,   // (T, B, 3)
    const float* __restrict__ hidden,  // (1, B, 3)
    const float* __restrict__ W_ih,    // (9, 3)
    const float* __restrict__ W_hh,    // (9, 3)
    const float* __restrict__ b_ih,    // (9)
    const float* __restrict__ b_hh,    // (9)
    float* __restrict__ out,           // (T,B,3) then (1,B,3), concatenated
    int T, int B)
{
    const int lane = threadIdx.x & 31;
    const int wave = (int)((blockIdx.x * blockDim.x + threadIdx.x) >> 5);
    const bool lo  = (lane < 16);

    int bg = wave * 16 + (lane & 15);    // batch column this lane serves
    const bool valid = (bg < B);
    if (bg >= B) bg = B - 1;             // clamp addresses; stores predicated

    // ---- A operand: W_hh as 16x4 (M = gate row 0..8, K = hidden 0..2).
    // 32-bit A layout: v0 = K0 (lanes 0-15) / K2 (lanes 16-31); v1 = K1 / K3.
    int g = lane & 15;
    float a0 = 0.0f, a1 = 0.0f;
    if (g < GG) {
        a0 = W_hh[g * 3 + (lo ? 0 : 2)];
        if (lo) a1 = W_hh[g * 3 + 1];
    }
    v2f A; A.x = a0; A.y = a1;

    // ---- C operand: b_hh broadcast along N per gate row M.
    // C/D layout: vgpr r -> M=r (lanes 0-15), M=r+8 (lanes 16-31).
    v8f C;
#pragma unroll
    for (int r = 0; r < 8; ++r) {
        int m = lo ? r : r + 8;
        C[r] = (m < GG) ? b_hh[m] : 0.0f;
    }

    // ---- per-lane copies of W_ih / b_ih for the x-side projection
    float wih[GG * HH], bih[GG];
#pragma unroll
    for (int i = 0; i < GG * HH; ++i) wih[i] = W_ih[i];
#pragma unroll
    for (int i = 0; i < GG; ++i) bih[i] = b_ih[i];

    // ---- initial hidden state, split ownership:
    // lanes 0-15 keep h0,h1; lanes 16-31 keep h2 (same batch column bg).
    float h0  = hidden[(size_t)bg * 3 + 0];
    float h1  = hidden[(size_t)bg * 3 + 1];
    float h2u = hidden[(size_t)bg * 3 + 2];   // meaningful in lanes 16-31

    const size_t strideT = (size_t)B * 3;
    const float* xp = input + (size_t)bg * 3;

    // register double-buffer for the streamed input
    float xn0 = xp[0], xn1 = xp[1], xn2 = xp[2];

    for (int t = 0; t < T; ++t) {
        float x0 = xn0, x1 = xn1, x2 = xn2;
        if (t + 1 < T) {                       // issue next-step load early
            const float* p1 = xp + (size_t)(t + 1) * strideT;
            xn0 = p1[0]; xn1 = p1[1]; xn2 = p1[2];
        }
        if (t + 32 < T)                        // deep prefetch: global_prefetch_b8
            __builtin_prefetch(xp + (size_t)(t + 32) * strideT, 0, 1);

        // ---- x-side projection: xg = b_ih + W_ih * x  (off the h-chain)
        float xg[GG];
#pragma unroll
        for (int q = 0; q < GG; ++q)
            xg[q] = bih[q] + wih[q * 3 + 0] * x0
                           + wih[q * 3 + 1] * x1
                           + wih[q * 3 + 2] * x2;

        // ---- h-side projection via WMMA: D = W_hh(16x4) * h^T(4x16) + b_hh
        // B operand needs h0(K0)/h1(K1) in lanes 0-15 and h2(K2) in lanes
        // 16-31 -- which is exactly where each component already lives.
        v2f Bm;
        Bm.x = lo ? h0 : h2u;
        Bm.y = lo ? h1 : 0.0f;

        v8f D = __builtin_amdgcn_wmma_f32_16x16x4_f32(
            false, A, false, Bm, (short)0, C, false, false);

        // ---- gate math.
        // Lanes 0-15: gates 0-7 are in-lane (D[0..7]); update h0,h1.
        float r0 = fast_sigmoid(xg[0] + D[0]);
        float r1 = fast_sigmoid(xg[1] + D[1]);
        float r2 = fast_sigmoid(xg[2] + D[2]);
        float z0 = fast_sigmoid(xg[3] + D[3]);
        float z1 = fast_sigmoid(xg[4] + D[4]);
        float z2 = fast_sigmoid(xg[5] + D[5]);

        float n0 = fast_tanh(xg[6] + r0 * D[6]);
        float n1 = fast_tanh(xg[7] + r1 * D[7]);
        h0 = n0 + z0 * (h0 - n0);
        h1 = n1 + z1 * (h1 - n1);

        // Lanes 16-31: gate 8 is in-lane (D[0] upper half); pull r2,z2 from
        // the lower half with VALU row-swap (no LDS pipe on the chain).
        float r2s = bcast_lo_to_hi(r2);        // upper lanes: r2 of batch bg
        float z2s = bcast_lo_to_hi(z2);        // upper lanes: z2 of batch bg
        float n2  = fast_tanh(xg[8] + r2s * D[0]);
        h2u = n2 + z2s * (h2u - n2);

        // ---- stores: component-split, every lane stores useful data
        if (valid) {
            float* op = out + (size_t)t * strideT + (size_t)bg * 3;
            if (lo) {
                op[0] = h0;
                op[1] = h1;
            } else {
                op[2] = h2u;
            }
        }   // reconverges before next WMMA (EXEC all-ones at WMMA)
    }

    // final hidden state appended after the (T,B,3) output block
    if (valid) {
        float* hp = out + (size_t)T * strideT + (size_t)bg * 3;
        if (lo) {
            hp[0] = h0;
            hp[1] = h1;
        } else {
            hp[2] = h2u;
        }
    }
}

extern "C" void kernel_launch(void* const* d_in, const int* in_sizes, int n_in,
                              void* d_out, int out_size, void* d_ws, size_t ws_size,
                              hipStream_t stream) {
    const float* input  = (const float*)d_in[0];
    const float* hidden = (const float*)d_in[1];
    const float* W_ih   = (const float*)d_in[2];
    const float* W_hh   = (const float*)d_in[3];
    const float* b_ih   = (const float*)d_in[4];
    const float* b_hh   = (const float*)d_in[5];
    float* out = (float*)d_out;

    const int B = in_sizes[1] / 3;            // hidden is (1,B,3)
    const int T = in_sizes[0] / (B * 3);      // input is (T,B,3)

    const int waves   = (B + 15) / 16;        // 16 batch columns per wave
    const int threads = 128;                  // 4 waves per block
    const int blocks  = (waves * 32 + threads - 1) / threads;

    gru_wmma_kernel<<<blocks, threads, 0, stream>>>(
        input, hidden, W_ih, W_hh, b_ih, b_hh, out, T, B);
}